// ANPM_81827716923970
// MI455X (gfx1250) — compile-verified
//
#include <hip/hip_runtime.h>
#include <hip/hip_bf16.h>
#include <math.h>

// ---------------------------------------------------------------------------
// GSimNet-style pipeline for MI455X (gfx1250, wave32):
//   attention (3 streaming passes over x, bandwidth-bound)
//   -> NTN bilinear via V_WMMA_F32_16X16X4_F32 (exact f32 dtype match)
//   -> tiny sigmoid MLP
// Deterministic: fixed-tree reductions through workspace, no float atomics.
// ---------------------------------------------------------------------------

#define D 256
#define DIN 512
#define NF 16

typedef float v2f __attribute__((ext_vector_type(2)));
typedef float v8f __attribute__((ext_vector_type(8)));

__device__ __forceinline__ float sigmoidf(float s) {
    return 1.0f / (1.0f + __expf(-s));
}

// ---- Pass A: column sums of x (partial per block) --------------------------
__global__ void k_colsum(const float* __restrict__ x, int N,
                         float* __restrict__ part) {
    int tid = threadIdx.x;                       // column 0..255
    float acc = 0.0f;
    for (int row = blockIdx.x; row < N; row += gridDim.x)
        acc += x[(size_t)row * D + tid];
    part[(size_t)blockIdx.x * D + tid] = acc;    // always written (0xAA-safe)
}

// ---- Reduce partials -> temp = sum*invN ; h = tanh(temp @ W0) --------------
__global__ void k_reduce_h(const float* __restrict__ part, int nPart, float invN,
                           const float* __restrict__ W0, float* __restrict__ h) {
    __shared__ float t[D];
    int tid = threadIdx.x;
    float s = 0.0f;
    for (int b = 0; b < nPart; ++b) s += part[(size_t)b * D + tid];
    t[tid] = s * invN;
    __syncthreads();
    float acc = 0.0f;
    for (int d = 0; d < D; ++d) acc += t[d] * W0[d * D + tid];
    h[tid] = tanhf(acc);
}

// ---- Attention streaming pass (one wave per row-slice, float4 loads) -------
// pass2==0:  att1[row] = sigmoid(x_row . h);      acc += att1 * x_row
// pass2==1:  att2 = sigmoid(att1*(x_row . h));    acc += att2*att1 * x_row
__global__ void k_att_pass(const float* __restrict__ x, int N,
                           const float* __restrict__ h,
                           float* __restrict__ att1,
                           float* __restrict__ part, int pass2) {
    int lane = threadIdx.x & 31;
    int wv   = threadIdx.x >> 5;
    int w    = blockIdx.x * (blockDim.x >> 5) + wv;
    int nw   = gridDim.x * (blockDim.x >> 5);

    const float4* h4 = (const float4*)h;         // 8 cols per lane
    float4 ha = h4[lane * 2 + 0];
    float4 hb = h4[lane * 2 + 1];

    float4 a0 = make_float4(0.f, 0.f, 0.f, 0.f);
    float4 a1 = make_float4(0.f, 0.f, 0.f, 0.f);

    for (int row = w; row < N; row += nw) {
        const float4* x4 = (const float4*)(x + (size_t)row * D);
        float4 xa = x4[lane * 2 + 0];
        float4 xb = x4[lane * 2 + 1];
        float s = xa.x * ha.x + xa.y * ha.y + xa.z * ha.z + xa.w * ha.w
                + xb.x * hb.x + xb.y * hb.y + xb.z * hb.z + xb.w * hb.w;
        #pragma unroll
        for (int o = 16; o; o >>= 1) s += __shfl_xor(s, o, 32);

        float wgt;
        if (pass2) {
            float a1v  = att1[row];
            float att2 = sigmoidf(a1v * s);
            wgt = att2 * a1v;                    // out2 = sum att2*att1*x
        } else {
            wgt = sigmoidf(s);
            if (lane == 0) att1[row] = wgt;
        }
        a0.x += wgt * xa.x; a0.y += wgt * xa.y; a0.z += wgt * xa.z; a0.w += wgt * xa.w;
        a1.x += wgt * xb.x; a1.y += wgt * xb.y; a1.z += wgt * xb.z; a1.w += wgt * xb.w;
    }
    float4* p4 = (float4*)(part + (size_t)w * D);
    p4[lane * 2 + 0] = a0;                       // always written (0xAA-safe)
    p4[lane * 2 + 1] = a1;
}

// ---- Reduce attention partials -> graph embedding g = [out2, out2] ---------
__global__ void k_reduce_g(const float* __restrict__ part, int nPart,
                           float* __restrict__ g) {
    int tid = threadIdx.x;
    float s = 0.0f;
    for (int b = 0; b < nPart; ++b) s += part[(size_t)b * D + tid];
    g[tid]     = s;                              // head 0
    g[D + tid] = s;                              // head 1 (identical params)
}

// ---- NTN bilinear term via V_WMMA_F32_16X16X4_F32 --------------------------
// One wave per (f, e-tile of 16). A = g1[d0:d0+4] broadcast to all 16 rows,
// B = W[f, d0:d0+4, e0:e0+16]. Accumulate over all d -> every row of C holds
// U_f[e0+n] = sum_d g1[d]*W[f,d,e0+n]; then dot with g2 and wave-reduce.
// Layout assumption (both A and B): K = vgpr_index + 2*(lane>=16).
__global__ void k_bilinear_wmma(const float* __restrict__ W,
                                const float* __restrict__ g1,
                                const float* __restrict__ g2,
                                float* __restrict__ wpart) {
    int f    = blockIdx.x >> 5;                  // 0..15
    int et   = blockIdx.x & 31;                  // 0..31
    int e0   = et * 16;
    int lane = threadIdx.x;                      // blockDim = 32, EXEC all-1s
    int ln   = lane & 15;
    int koff = (lane >> 4) * 2;                  // 0 for lanes 0-15, 2 for 16-31

    const float* Wf = W + (size_t)f * (DIN * DIN);

    v8f c = {};
    for (int d0 = 0; d0 < DIN; d0 += 4) {
        v2f a, b;
        a.x = g1[d0 + koff];
        a.y = g1[d0 + koff + 1];
        b.x = Wf[(size_t)(d0 + koff)     * DIN + e0 + ln];
        b.y = Wf[(size_t)(d0 + koff + 1) * DIN + e0 + ln];
        c = __builtin_amdgcn_wmma_f32_16x16x4_f32(
                /*neg_a=*/false, a, /*neg_b=*/false, b,
                /*c_mod=*/(short)0, c, /*reuse_a=*/false, /*reuse_b=*/false);
    }
    // Row M=0 of D lives in c[0] on lanes 0..15 (N = lane).
    float u = c[0];
    float v = (lane < 16) ? u * g2[e0 + lane] : 0.0f;
    #pragma unroll
    for (int o = 16; o; o >>= 1) v += __shfl_xor(v, o, 32);
    if (lane == 0) wpart[f * 32 + et] = v;
}

// ---- NTN combine (v-term + w-term + bias, sigmoid) + 16->8->4->2->1 MLP ----
__global__ void k_ntn_mlp(const float* __restrict__ V, const float* __restrict__ b,
                          const float* __restrict__ g1, const float* __restrict__ g2,
                          const float* __restrict__ wpart,
                          const float* __restrict__ P0, const float* __restrict__ P1,
                          const float* __restrict__ P2, const float* __restrict__ P3,
                          float* __restrict__ out) {
    int lane = threadIdx.x;                      // 32 threads
    __shared__ float sc[NF];
    for (int f = 0; f < NF; ++f) {
        float vs = 0.0f;
        for (int k = lane; k < 2 * DIN; k += 32) {
            float cv = (k < DIN) ? g1[k] : g2[k - DIN];
            vs += cv * V[f * (2 * DIN) + k];
        }
        float t = vs + wpart[f * 32 + lane];     // exactly 32 bilinear partials
        #pragma unroll
        for (int o = 16; o; o >>= 1) t += __shfl_xor(t, o, 32);
        if (lane == 0) sc[f] = sigmoidf(t + b[f]);
    }
    __syncthreads();
    if (lane == 0) {
        float h0[8], h1v[4], h2v[2];
        for (int o = 0; o < 8; ++o) {
            float s = 0.f;
            for (int j = 0; j < 16; ++j) s += P0[o * 16 + j] * sc[j];
            h0[o] = sigmoidf(s);
        }
        for (int o = 0; o < 4; ++o) {
            float s = 0.f;
            for (int j = 0; j < 8; ++j) s += P1[o * 8 + j] * h0[j];
            h1v[o] = sigmoidf(s);
        }
        for (int o = 0; o < 2; ++o) {
            float s = 0.f;
            for (int j = 0; j < 4; ++j) s += P2[o * 4 + j] * h1v[j];
            h2v[o] = sigmoidf(s);
        }
        float s = P3[0] * h2v[0] + P3[1] * h2v[1];
        out[0] = sigmoidf(s);
    }
}

// ---------------------------------------------------------------------------
static void run_attention(const float* x, int N, const float* W0, float* G,
                          float* H, float* CPART, float* APART, float* ATT1,
                          hipStream_t stream) {
    const int CS_BLOCKS  = 256;                  // colsum partial vectors
    const int ATT_BLOCKS = 120;                  // 120 blocks * 8 waves = 960
    const int ATT_WAVES  = ATT_BLOCKS * 8;
    float invN = 1.0f / (float)N;

    k_colsum  <<<CS_BLOCKS, 256, 0, stream>>>(x, N, CPART);
    k_reduce_h<<<1, 256, 0, stream>>>(CPART, CS_BLOCKS, invN, W0, H);   // h1
    k_att_pass<<<ATT_BLOCKS, 256, 0, stream>>>(x, N, H, ATT1, APART, 0);
    k_reduce_h<<<1, 256, 0, stream>>>(APART, ATT_WAVES, invN, W0, H);   // h2
    k_att_pass<<<ATT_BLOCKS, 256, 0, stream>>>(x, N, H, ATT1, APART, 1);
    k_reduce_g<<<1, 256, 0, stream>>>(APART, ATT_WAVES, G);
}

extern "C" void kernel_launch(void* const* d_in, const int* in_sizes, int n_in,
                              void* d_out, int out_size, void* d_ws, size_t ws_size,
                              hipStream_t stream) {
    const float* x1 = (const float*)d_in[0];
    const float* x2 = (const float*)d_in[1];
    const float* W0 = (const float*)d_in[2];
    const float* V  = (const float*)d_in[3];
    const float* W  = (const float*)d_in[4];
    const float* b  = (const float*)d_in[5];
    const float* P0 = (const float*)d_in[6];
    const float* P1 = (const float*)d_in[7];
    const float* P2 = (const float*)d_in[8];
    const float* P3 = (const float*)d_in[9];
    float* out = (float*)d_out;

    int N1 = in_sizes[0] / D;
    int N2 = in_sizes[1] / D;

    // Workspace layout (floats):
    float* ws    = (float*)d_ws;
    float* G1    = ws;                           // 512
    float* G2    = ws + 512;                     // 512
    float* H     = ws + 1024;                    // 256
    float* WPART = ws + 1280;                    // 512 (16 f * 32 e-tiles)
    float* CPART = ws + 2048;                    // 256 * 256
    float* APART = CPART + 256 * 256;            // 960 * 256
    float* ATT1  = APART + 960 * 256;            // max(N1, N2)

    run_attention(x1, N1, W0, G1, H, CPART, APART, ATT1, stream);
    run_attention(x2, N2, W0, G2, H, CPART, APART, ATT1, stream);

    k_bilinear_wmma<<<NF * 32, 32, 0, stream>>>(W, G1, G2, WPART);
    k_ntn_mlp<<<1, 32, 0, stream>>>(V, b, G1, G2, WPART,
                                    P0, P1, P2, P3, out);
}